// MinibatchDiscrimination_14757507629813
// MI455X (gfx1250) — compile-verified
//
#include <hip/hip_runtime.h>
#include <hip/hip_bf16.h>

typedef float v2f __attribute__((ext_vector_type(2)));
typedef float v8f __attribute__((ext_vector_type(8)));

#define BB 512   // batch
#define FF 512   // features
#define KK 32    // num_kernels
#define DD 16    // dim_per_kernel
#define NN (KK * DD)     // 512 = cols of m
#define OUTC (FF + KK)   // 544 = output row width

// ---------------------------------------------------------------------------
// m = x @ T  via V_WMMA_F32_16X16X4_F32 (exact fp32 matrix-core path).
// One wave computes one 16x16 tile of m. 8 waves per block, 1024 tiles total.
// A-fragment lane layout (ISA 7.12.2, 32-bit A 16x4):
//   lane l (l<16):  {A[l][k], A[l][k+1]};  lane l (>=16): {A[l-16][k+2], A[l-16][k+3]}
// B-fragment (4x16, rows striped across half-waves symmetric to A):
//   lanes 0-15 hold rows k,k+1 at col=lane; lanes 16-31 hold rows k+2,k+3.
// C/D: VGPR r -> row (r + 8*(lane>=16)), col = lane%16.
// ---------------------------------------------------------------------------
__global__ __launch_bounds__(256) void gemm_wmma_f32(const float* __restrict__ x,
                                                     const float* __restrict__ T,
                                                     float* __restrict__ m) {
  const int lane = threadIdx.x & 31;
  const int wave = threadIdx.x >> 5;
  const int tile = blockIdx.x * 8 + wave;   // 0..1023
  const int tile_m = tile >> 5;             // 32 tiles of rows
  const int tile_n = tile & 31;             // 32 tiles of cols
  const int half = lane >> 4;               // 0 or 1
  const int l16 = lane & 15;

  const int arow = tile_m * 16 + l16;       // row of x this lane loads
  const int bcol = tile_n * 16 + l16;       // col of T this lane loads

  v8f c = {};
  for (int k = 0; k < FF; k += 4) {
    const int ka = k + half * 2;
    v2f a;
    a.x = x[arow * FF + ka];
    a.y = x[arow * FF + ka + 1];
    v2f b;
    b.x = T[(ka)     * NN + bcol];
    b.y = T[(ka + 1) * NN + bcol];
    c = __builtin_amdgcn_wmma_f32_16x16x4_f32(false, a, false, b,
                                              (short)0, c, false, false);
  }

#pragma unroll
  for (int r = 0; r < 8; ++r) {
    m[(tile_m * 16 + r + half * 8) * NN + bcol] = c[r];
  }
}

// ---------------------------------------------------------------------------
// out[:, :512] = x
// ---------------------------------------------------------------------------
__global__ __launch_bounds__(256) void copy_x(const float* __restrict__ x,
                                              float* __restrict__ out) {
  const int idx = blockIdx.x * 256 + threadIdx.x;
  if (idx < BB * FF) {
    const int i = idx / FF;
    const int ccol = idx - i * FF;
    out[i * OUTC + ccol] = x[idx];
  }
}

// ---------------------------------------------------------------------------
// feats[i][k] = sum_j exp( -sum_d |m[i,k,d] - m[j,k,d]| )
// Block: (i-tile of 64, one k). Stage m[:,k,:] (512x16 fp32 = 32KB) into LDS
// once; j-loop then reads LDS with full-wave broadcast (all lanes same addr).
// ---------------------------------------------------------------------------
__global__ __launch_bounds__(64) void feats_kernel(const float* __restrict__ m,
                                                   float* __restrict__ out) {
  __shared__ float lds[BB * DD];  // 32 KB
  const int k = blockIdx.y;
  const int i = blockIdx.x * 64 + threadIdx.x;

  // cooperative stage: 512 rows x 4 float4 each = 2048 float4 loads
  for (int t = threadIdx.x; t < BB * 4; t += 64) {
    const int j = t >> 2;
    const int q = t & 3;
    const float4* src = (const float4*)(m + j * NN + k * DD);
    ((float4*)lds)[j * 4 + q] = src[q];
  }
  __syncthreads();

  float4 mi0 = ((const float4*)(lds + i * DD))[0];
  float4 mi1 = ((const float4*)(lds + i * DD))[1];
  float4 mi2 = ((const float4*)(lds + i * DD))[2];
  float4 mi3 = ((const float4*)(lds + i * DD))[3];

  float acc = 0.f;
  for (int j = 0; j < BB; ++j) {
    const float4* row = (const float4*)(lds + j * DD);
    float4 r0 = row[0], r1 = row[1], r2 = row[2], r3 = row[3];
    float dist = fabsf(mi0.x - r0.x) + fabsf(mi0.y - r0.y) +
                 fabsf(mi0.z - r0.z) + fabsf(mi0.w - r0.w);
    dist += fabsf(mi1.x - r1.x) + fabsf(mi1.y - r1.y) +
            fabsf(mi1.z - r1.z) + fabsf(mi1.w - r1.w);
    dist += fabsf(mi2.x - r2.x) + fabsf(mi2.y - r2.y) +
            fabsf(mi2.z - r2.z) + fabsf(mi2.w - r2.w);
    dist += fabsf(mi3.x - r3.x) + fabsf(mi3.y - r3.y) +
            fabsf(mi3.z - r3.z) + fabsf(mi3.w - r3.w);
    acc += __expf(-dist);
  }
  out[i * OUTC + FF + k] = acc;
}

// ---------------------------------------------------------------------------
extern "C" void kernel_launch(void* const* d_in, const int* in_sizes, int n_in,
                              void* d_out, int out_size, void* d_ws, size_t ws_size,
                              hipStream_t stream) {
  (void)in_sizes; (void)n_in; (void)out_size; (void)ws_size;
  const float* x = (const float*)d_in[0];
  const float* T = (const float*)d_in[1];
  float* out = (float*)d_out;
  float* m = (float*)d_ws;  // [512 x 512] fp32 scratch = 1 MB

  gemm_wmma_f32<<<128, 256, 0, stream>>>(x, T, m);             // 1024 tiles / 8 waves
  copy_x<<<(BB * FF + 255) / 256, 256, 0, stream>>>(x, out);
  feats_kernel<<<dim3(8, 32), 64, 0, stream>>>(m, out);
}